// lut_silu_kb_18863496364228
// MI455X (gfx1250) — compile-verified
//
#include <hip/hip_runtime.h>

// ---------------------------------------------------------------------------
// LUT-SiLU (k*x+b piecewise-linear fp16 LUT), MI455X / gfx1250.
// Bandwidth-bound: 8 B/elem -> ~537 MB @ 23.3 TB/s ~= 23 us floor.
// Index math done as fp16 bit surgery; LUT staged to LDS via TDM when the
// toolchain exposes __builtin_amdgcn_tensor_load_to_lds.
// ---------------------------------------------------------------------------

typedef _Float16     half2_t __attribute__((ext_vector_type(2)));
typedef unsigned int u32x4   __attribute__((ext_vector_type(4)));
typedef int          i32x8   __attribute__((ext_vector_type(8)));
typedef int          i32x4   __attribute__((ext_vector_type(4)));

#define LUT_N 1024   // 2 (sign) * 32 (levels) * 16 (entries)

#if __has_builtin(__builtin_amdgcn_tensor_load_to_lds) && \
    __has_builtin(__builtin_amdgcn_s_wait_tensorcnt)
#define USE_TDM 1
#endif

// fp16 bits -> LUT index, exactly mirroring the reference's fp32-math
// extraction:
//   normal    (E in 1..30): lvl=E,            mi=M>>6
//   subnormal (E=0, M!=0) : lvl=1,            mi=0   (log2 clip to -14, round->clip 0)
//   zero      (E=0, M=0)  : lvl=0, mi=0, sign forced 0 (reference uses xf<0)
//   inf       (E=31,M=0)  : lvl=31,           mi=0
//   nan       (E=31,M!=0) : index is don't-care (result overridden)
__device__ __forceinline__ unsigned lut_index(unsigned u) {
  unsigned m = u & 0x3FFu;
  unsigned E = (u >> 10) & 0x1Fu;
  unsigned s = (u >> 15) & 1u;
  s &= ((u & 0x7FFFu) != 0u);                                   // +/-0 -> sign 0
  unsigned lm = (E == 0u) ? (m ? 16u : 0u) : ((E << 4) | (m >> 6));
  return (s << 9) | lm;                                         // < 1024 always
}

#ifdef USE_TDM
// 1D TDM copy: `bytes` (multiple of 4) from global to LDS, dword elements.
// D# group0: count=1 | lds_addr | global_addr[31:0] | ga[56:32] + type=2.
// D# group1: data_size=4B, tensor_dim0=tile_dim0=dwords, dim1=1, stride0=dwords.
__device__ __forceinline__ void tdm_load_1d(const void* g, void* l, unsigned bytes) {
  unsigned long long ga = (unsigned long long)(size_t)g;
  unsigned la = (unsigned)(size_t)l;   // generic LDS ptr: low 32 bits = LDS byte addr
  unsigned dw = bytes >> 2;
  u32x4 g0 = { 1u,                                              // count=1 (valid user D#)
               la,                                              // lds_addr
               (unsigned)ga,                                    // global_addr[31:0]
               (unsigned)(((ga >> 32) & 0x1FFFFFFull) | 0x80000000u) }; // ga[56:32] | type=2
  i32x8 g1 = { (int)(2u << 16),                                 // wg_mask=0, data_size=4B
               (int)((dw & 0xFFFFu) << 16),                     // tensor_dim0[15:0]
               (int)(((dw >> 16) & 0xFFFFu) | (1u << 16)),      // tensor_dim0[31:16], tensor_dim1=1 (lo)
               (int)((dw & 0xFFFFu) << 16),                     // tensor_dim1 hi = 0, tile_dim0 = dw
               0,                                               // tile_dim1=0, tile_dim2=0
               (int)dw,                                         // tensor_dim0_stride[31:0]
               0, 0 };
  i32x4 g2 = {0, 0, 0, 0};
  i32x4 g3 = {0, 0, 0, 0};
#if defined(__clang_major__) && (__clang_major__ >= 23)
  i32x8 g4 = {0, 0, 0, 0, 0, 0, 0, 0};
  __builtin_amdgcn_tensor_load_to_lds(g0, g1, g2, g3, g4, 0);
#else
  __builtin_amdgcn_tensor_load_to_lds(g0, g1, g2, g3, 0);
#endif
}
#endif // USE_TDM

__global__ void __launch_bounds__(256)
lut_silu_kb_main(const float4* __restrict__ x4,
                 const unsigned short* __restrict__ kt,
                 const unsigned short* __restrict__ bt,
                 float4* __restrict__ out4, int nvec) {
  __shared__ unsigned lut[LUT_N];   // interleaved {b[31:16], k[15:0]}

#ifdef USE_TDM
  __shared__ unsigned short lk[LUT_N];
  __shared__ unsigned short lb[LUT_N];
  if (threadIdx.x < 32u) {          // wave 0 only; TDM ignores EXEC, per-wave issue
    tdm_load_1d(kt, lk, LUT_N * 2u);
    tdm_load_1d(bt, lb, LUT_N * 2u);
    __builtin_amdgcn_s_wait_tensorcnt(0);
  }
  __syncthreads();
  for (unsigned j = threadIdx.x; j < LUT_N; j += blockDim.x)
    lut[j] = (unsigned)lk[j] | ((unsigned)lb[j] << 16);
#else
  for (unsigned j = threadIdx.x; j < LUT_N; j += blockDim.x)
    lut[j] = (unsigned)kt[j] | ((unsigned)bt[j] << 16);
#endif
  __syncthreads();

  int tid    = (int)(blockIdx.x * blockDim.x + threadIdx.x);
  int stride = (int)(gridDim.x * blockDim.x);

  for (int i = tid; i < nvec; i += stride) {
    if (i + stride < nvec)
      __builtin_prefetch(&x4[i + stride], 0, 0);   // -> global_prefetch_b8

    float4 v = x4[i];                               // global_load_b128

    _Float16 ha = (_Float16)v.x, hb = (_Float16)v.y;
    _Float16 hc = (_Float16)v.z, hd = (_Float16)v.w;
    unsigned u0 = (unsigned)__builtin_bit_cast(unsigned short, ha);
    unsigned u1 = (unsigned)__builtin_bit_cast(unsigned short, hb);
    unsigned u2 = (unsigned)__builtin_bit_cast(unsigned short, hc);
    unsigned u3 = (unsigned)__builtin_bit_cast(unsigned short, hd);

    unsigned kb0 = lut[lut_index(u0)];              // ds_load_b32 x4
    unsigned kb1 = lut[lut_index(u1)];
    unsigned kb2 = lut[lut_index(u2)];
    unsigned kb3 = lut[lut_index(u3)];

    unsigned k01 = (kb0 & 0xFFFFu) | (kb1 << 16);
    unsigned b01 = (kb0 >> 16)     | (kb1 & 0xFFFF0000u);
    unsigned k23 = (kb2 & 0xFFFFu) | (kb3 << 16);
    unsigned b23 = (kb2 >> 16)     | (kb3 & 0xFFFF0000u);

    half2_t x01 = {ha, hb};
    half2_t x23 = {hc, hd};
    half2_t r01 = __builtin_elementwise_fma(__builtin_bit_cast(half2_t, k01), x01,
                                            __builtin_bit_cast(half2_t, b01));  // v_pk_fma_f16
    half2_t r23 = __builtin_elementwise_fma(__builtin_bit_cast(half2_t, k23), x23,
                                            __builtin_bit_cast(half2_t, b23));

    // NaN propagation: fp16 NaN <=> (bits & 0x7FFF) > 0x7C00; force 0x7E00.
    unsigned rr01 = __builtin_bit_cast(unsigned, r01);
    unsigned rr23 = __builtin_bit_cast(unsigned, r23);
    rr01 = ((u0 & 0x7FFFu) > 0x7C00u) ? ((rr01 & 0xFFFF0000u) | 0x00007E00u) : rr01;
    rr01 = ((u1 & 0x7FFFu) > 0x7C00u) ? ((rr01 & 0x0000FFFFu) | 0x7E000000u) : rr01;
    rr23 = ((u2 & 0x7FFFu) > 0x7C00u) ? ((rr23 & 0xFFFF0000u) | 0x00007E00u) : rr23;
    rr23 = ((u3 & 0x7FFFu) > 0x7C00u) ? ((rr23 & 0x0000FFFFu) | 0x7E000000u) : rr23;
    half2_t f01 = __builtin_bit_cast(half2_t, rr01);
    half2_t f23 = __builtin_bit_cast(half2_t, rr23);

    float4 o;
    o.x = (float)f01.x; o.y = (float)f01.y;
    o.z = (float)f23.x; o.w = (float)f23.y;
    out4[i] = o;                                    // global_store_b128
  }
}

// Scalar tail for n % 4 != 0 (not hit for 4*4096*4096, but keep it general).
__global__ void __launch_bounds__(64)
lut_silu_kb_tail(const float* __restrict__ x,
                 const unsigned short* __restrict__ kt,
                 const unsigned short* __restrict__ bt,
                 float* __restrict__ out, int start, int n) {
  int i = start + (int)(blockIdx.x * blockDim.x + threadIdx.x);
  if (i >= n) return;
  _Float16 h = (_Float16)x[i];
  unsigned u = (unsigned)__builtin_bit_cast(unsigned short, h);
  unsigned lin = lut_index(u);
  _Float16 k = __builtin_bit_cast(_Float16, (unsigned short)kt[lin]);
  _Float16 b = __builtin_bit_cast(_Float16, (unsigned short)bt[lin]);
  _Float16 r = (_Float16)__builtin_fmaf16(k, h, b);
  unsigned short rb = ((u & 0x7FFFu) > 0x7C00u) ? (unsigned short)0x7E00u
                                                : __builtin_bit_cast(unsigned short, r);
  out[i] = (float)__builtin_bit_cast(_Float16, rb);
}

extern "C" void kernel_launch(void* const* d_in, const int* in_sizes, int n_in,
                              void* d_out, int out_size, void* d_ws, size_t ws_size,
                              hipStream_t stream) {
  const float*          x  = (const float*)d_in[0];
  const unsigned short* kt = (const unsigned short*)d_in[1];  // fp16 raw bits
  const unsigned short* bt = (const unsigned short*)d_in[2];  // fp16 raw bits
  float*                out = (float*)d_out;

  int n    = in_sizes[0];
  int nvec = n >> 2;
  int rem  = n & 3;

  if (nvec > 0) {
    int blocks = (nvec + 255) / 256;
    if (blocks > 4096) blocks = 4096;   // grid-stride; ~16 vec4 iters/thread at N=67M
    lut_silu_kb_main<<<blocks, 256, 0, stream>>>(
        (const float4*)x, kt, bt, (float4*)out, nvec);
  }
  if (rem > 0) {
    lut_silu_kb_tail<<<1, 64, 0, stream>>>(x, kt, bt, out, nvec << 2, n);
  }
}